// LocalFeatureAggregation_84284438217245
// MI455X (gfx1250) — compile-verified
//
#include <hip/hip_runtime.h>
#include <math.h>

// ---------------- problem constants ----------------
#define BB   2
#define NNP  32768          // points per batch
#define KKN  16             // neighbors
#define DIN  8
#define DH   16             // d_out/2
#define DOUTC 32            // d_out
#define EPSB 1e-6f
#define NPTS (BB*NNP)       // 65536 total points

typedef float v2f __attribute__((ext_vector_type(2)));
typedef float v8f __attribute__((ext_vector_type(8)));

#if defined(__has_builtin)
# if __has_builtin(__builtin_amdgcn_wmma_f32_16x16x4_f32)
#  define HAVE_WMMA_F32X4 1
# endif
#endif
#ifndef HAVE_WMMA_F32X4
# define HAVE_WMMA_F32X4 0
#endif

// ======================================================================
// Kernel 1: brute-force KNN, per-thread query, LDS-staged candidate tiles.
// Register-resident top-16 kept sorted descending (bd[0] = current worst),
// fully unrolled sift so indices stay static -> no scratch spill.
// ======================================================================
#define KTILE 1024
__global__ __launch_bounds__(256) void knn_kernel(const float* __restrict__ coords,
                                                  int* __restrict__ idx) {
    __shared__ float4 tile[KTILE];                     // 16 KB
    const int tid = threadIdx.x;
    const int b   = blockIdx.x >> 7;                   // 128 blocks per batch
    const int n   = ((blockIdx.x & 127) << 8) + tid;
    const float* cb = coords + (size_t)b * NNP * 3;

    const float qx = cb[n*3+0], qy = cb[n*3+1], qz = cb[n*3+2];
    const float q2 = qx*qx + qy*qy + qz*qz;

    float bd[KKN]; int bi[KKN];
#pragma unroll
    for (int i = 0; i < KKN; ++i) { bd[i] = 3.4e38f; bi[i] = 0; }

    for (int t = 0; t < NNP / KTILE; ++t) {
        __syncthreads();
        for (int i = tid; i < KTILE; i += 256) {
            const int j = t * KTILE + i;
            const float cx = cb[j*3+0], cy = cb[j*3+1], cz = cb[j*3+2];
            tile[i] = make_float4(cx, cy, cz, cx*cx + cy*cy + cz*cz);
        }
        __syncthreads();
#pragma unroll 4
        for (int i = 0; i < KTILE; ++i) {
            const float4 c = tile[i];                  // LDS broadcast, no conflicts
            const float d = (q2 + c.w) - 2.0f * (qx*c.x + qy*c.y + qz*c.z);
            if (d < bd[0]) {                           // rare after warm-up
                bd[0] = d; bi[0] = t * KTILE + i;
#pragma unroll
                for (int s = 0; s < KKN-1; ++s) {      // sift to keep descending
                    if (bd[s] < bd[s+1]) {
                        float td = bd[s]; bd[s] = bd[s+1]; bd[s+1] = td;
                        int   ti = bi[s]; bi[s] = bi[s+1]; bi[s+1] = ti;
                    }
                }
            }
        }
    }
    int* op = idx + ((size_t)b * NNP + n) * KKN;
#pragma unroll
    for (int i = 0; i < KKN; ++i) op[i] = bi[i];
}

// ======================================================================
// Kernel 2: mlp1 (leaky 0.2) + shortcut pre-BN linear, one thread per point.
// features layout (B,8,N,1): feats[b][n][j] = features[(b*8+j)*N + n]
// ======================================================================
__global__ __launch_bounds__(256) void mlp1_sc_kernel(
        const float* __restrict__ features,
        const float* __restrict__ W1,  const float* __restrict__ b1,
        const float* __restrict__ Wsc, const float* __restrict__ bsc,
        float* __restrict__ x1, float* __restrict__ scpre) {
    const int p = blockIdx.x * 256 + threadIdx.x;
    const int b = p / NNP, n = p % NNP;
    float f[DIN];
#pragma unroll
    for (int j = 0; j < DIN; ++j) f[j] = features[((size_t)b*DIN + j)*NNP + n];
#pragma unroll
    for (int c = 0; c < DH; ++c) {
        float v = b1[c];
#pragma unroll
        for (int j = 0; j < DIN; ++j) v = fmaf(f[j], W1[c*DIN+j], v);
        x1[(size_t)p*DH + c] = v > 0.f ? v : 0.2f * v;
    }
    for (int c = 0; c < 2*DOUTC; ++c) {
        float v = bsc[c];
#pragma unroll
        for (int j = 0; j < DIN; ++j) v = fmaf(f[j], Wsc[c*DIN+j], v);
        scpre[(size_t)p*(2*DOUTC) + c] = v;
    }
}

// ======================================================================
// Kernel 3: BN stats of the LSE pre-activation h_pre (recomputed, never
// materialized). Deterministic block partials: partials[block][2*16].
// ======================================================================
#define LSG 1024
__global__ __launch_bounds__(256) void lse_stats_kernel(
        const float* __restrict__ coords, const int* __restrict__ idx,
        const float* __restrict__ Wl, const float* __restrict__ bl,
        float* __restrict__ partials) {
    float sum[DH], ssq[DH];
#pragma unroll
    for (int c = 0; c < DH; ++c) { sum[c] = 0.f; ssq[c] = 0.f; }
    const int S = NPTS * KKN;
    for (int s = blockIdx.x * 256 + threadIdx.x; s < S; s += LSG * 256) {
        const int pn = s / KKN;
        const int b = pn / NNP, n = pn % NNP;
        const float* cb = coords + (size_t)b * NNP * 3;
        const float cx = cb[n*3], cy = cb[n*3+1], cz = cb[n*3+2];
        const int j = idx[s];
        const float nx = cb[j*3], ny = cb[j*3+1], nz = cb[j*3+2];
        const float cat[10] = {cx, cy, cz, nx, ny, nz, cx-nx, cy-ny, cz-nz, 1.f};
#pragma unroll
        for (int c = 0; c < DH; ++c) {
            float v = bl[c];
#pragma unroll
            for (int q = 0; q < 10; ++q) v = fmaf(cat[q], Wl[c*10+q], v);
            sum[c] += v; ssq[c] = fmaf(v, v, ssq[c]);
        }
    }
    __shared__ float red[256];
    for (int q = 0; q < 2*DH; ++q) {
        red[threadIdx.x] = (q < DH) ? sum[q] : ssq[q-DH];
        __syncthreads();
        for (int off = 128; off > 0; off >>= 1) {
            if (threadIdx.x < off) red[threadIdx.x] += red[threadIdx.x + off];
            __syncthreads();
        }
        if (threadIdx.x == 0) partials[blockIdx.x*(2*DH) + q] = red[0];
        __syncthreads();
    }
}

// ======================================================================
// Kernel 4: generic BN stats over a materialized (S,C) matrix.
// thread owns a channel; deterministic in-block combine.
// ======================================================================
__global__ __launch_bounds__(256) void bn_stats_mat(const float* __restrict__ X,
                                                    int S, int C,
                                                    float* __restrict__ partials) {
    const int c = threadIdx.x % C;
    const int r = threadIdx.x / C;
    const int tpc = 256 / C;
    float sum = 0.f, ssq = 0.f;
    for (int s = blockIdx.x * tpc + r; s < S; s += gridDim.x * tpc) {
        const float v = X[(size_t)s * C + c];
        sum += v; ssq = fmaf(v, v, ssq);
    }
    __shared__ float sh[512];
    sh[threadIdx.x] = sum; sh[256 + threadIdx.x] = ssq;
    __syncthreads();
    if (threadIdx.x < C) {
        float ts = 0.f, tq = 0.f;
        for (int rr = 0; rr < tpc; ++rr) { ts += sh[rr*C + c]; tq += sh[256 + rr*C + c]; }
        partials[blockIdx.x*2*C + c]     = ts;
        partials[blockIdx.x*2*C + C + c] = tq;
    }
}

// mv[c] = mean, mv[C+c] = rsqrt(var+eps); ordered scan -> deterministic
__global__ void bn_finalize(const float* __restrict__ partials, int G, float count,
                            int C, float* __restrict__ mv) {
    const int c = threadIdx.x;
    if (c >= C) return;
    float s = 0.f, q = 0.f;
    for (int g = 0; g < G; ++g) { s += partials[g*2*C + c]; q += partials[g*2*C + C + c]; }
    const float m = s / count;
    const float var = q / count - m*m;
    mv[c] = m; mv[C + c] = rsqrtf(var + EPSB);
}

__global__ __launch_bounds__(256) void bn_relu_kernel(const float* __restrict__ X,
        const float* __restrict__ mv, const float* __restrict__ g,
        const float* __restrict__ bb, int S, int C, float* __restrict__ Y) {
    const int i = blockIdx.x * 256 + threadIdx.x;
    if (i >= S * C) return;
    const int c = i % C;
    const float v = (X[i] - mv[c]) * mv[C + c] * g[c] + bb[c];
    Y[i] = v > 0.f ? v : 0.f;
}

// ======================================================================
// Kernel 5: fused LSE (BN+relu applied with precomputed stats) + attention
// pool. One wave per point. Attention logits s = x_cat(16x32) @ Ws^T(32x32)
// computed with V_WMMA_F32_16X16X4_F32: 8 chained K=4 steps x 2 N-tiles.
// A layout (f32 16x4): lanes 0-15 -> rows M=0..15, K=0..1; lanes 16-31 ->
// same rows, K=2..3. C/D layout: VGPR r holds M=r (lanes 0-15) / M=r+8.
// ======================================================================
__global__ __launch_bounds__(256) void attpool_kernel(
        const float* __restrict__ coords, const int* __restrict__ idx,
        const float* __restrict__ xin,
        const float* __restrict__ Wl,  const float* __restrict__ bl,
        const float* __restrict__ mvl, const float* __restrict__ gl,
        const float* __restrict__ blb,
        const float* __restrict__ Ws,  const float* __restrict__ Wp,
        const float* __restrict__ bp,  int Cp, float* __restrict__ fpre) {
    __shared__ float sWs[DOUTC*DOUTC];     // 1024
    __shared__ float sWl[DH*10];           // 160
    __shared__ float sWp[DOUTC*DOUTC];     // up to 1024
    __shared__ float sPar[4*DH + DOUTC];   // mean, inv, gamma, beta, bp
    __shared__ float sX[8][KKN][DOUTC+1];  // x_cat per wave, padded stride 33
    __shared__ float sS[8][KKN][DOUTC+1];  // logits per wave
    __shared__ float sF[8][DOUTC];

    const int tid = threadIdx.x;
    for (int i = tid; i < DOUTC*DOUTC; i += 256) sWs[i] = Ws[i];
    for (int i = tid; i < DH*10;       i += 256) sWl[i] = Wl[i];
    for (int i = tid; i < Cp*DOUTC;    i += 256) sWp[i] = Wp[i];
    if (tid < DH) { sPar[tid] = mvl[tid]; sPar[DH+tid] = mvl[DH+tid];
                    sPar[2*DH+tid] = gl[tid]; sPar[3*DH+tid] = blb[tid]; }
    if (tid < Cp) sPar[4*DH + tid] = bp[tid];
    __syncthreads();

    const int w    = tid >> 5;
    const int lane = tid & 31;
    const int pt   = blockIdx.x * 8 + w;
    const int b    = pt / NNP, n = pt % NNP;
    const float* cb = coords + (size_t)b * NNP * 3;
    const int k    = lane & 15;
    const int half = lane >> 4;

    // ---- step 1: build x_cat = [relu(BN(lse_linear)) (16) | xin (16)] ----
    {
        const float cx = cb[n*3], cy = cb[n*3+1], cz = cb[n*3+2];
        const int j = idx[(size_t)pt*KKN + k];
        const float nx = cb[j*3], ny = cb[j*3+1], nz = cb[j*3+2];
        const float cat[10] = {cx, cy, cz, nx, ny, nz, cx-nx, cy-ny, cz-nz, 1.f};
#pragma unroll
        for (int q = 0; q < 8; ++q) {
            const int c = half*8 + q;
            float v = bl[c];
#pragma unroll
            for (int u = 0; u < 10; ++u) v = fmaf(cat[u], sWl[c*10+u], v);
            v = (v - sPar[c]) * sPar[DH+c] * sPar[2*DH+c] + sPar[3*DH+c];
            sX[w][k][c] = v > 0.f ? v : 0.f;
        }
#pragma unroll
        for (int q = 0; q < 8; ++q) {
            const int c = half*8 + q;
            sX[w][k][DH + c] = xin[(size_t)pt*DH + c];   // broadcast over k
        }
    }
    // same-wave LDS ops are in order; no cross-wave sharing of sX/sS/sF.

    // ---- step 2: s = x_cat @ Ws^T via WMMA f32 16x16x4 ----
#if HAVE_WMMA_F32X4
    {
        v8f acc0 = {0.f,0.f,0.f,0.f,0.f,0.f,0.f,0.f};
        v8f acc1 = {0.f,0.f,0.f,0.f,0.f,0.f,0.f,0.f};
#pragma unroll
        for (int cc = 0; cc < DOUTC; cc += 4) {
            const int co = cc + half*2;                  // this half's K-slice
            v2f A;  A[0]  = sX[w][k][co];          A[1]  = sX[w][k][co+1];
            v2f B0; B0[0] = sWs[k*DOUTC + co];     B0[1] = sWs[k*DOUTC + co + 1];
            v2f B1; B1[0] = sWs[(16+k)*DOUTC+co];  B1[1] = sWs[(16+k)*DOUTC+co+1];
            acc0 = __builtin_amdgcn_wmma_f32_16x16x4_f32(false, A, false, B0,
                                                         (short)0, acc0, false, false);
            acc1 = __builtin_amdgcn_wmma_f32_16x16x4_f32(false, A, false, B1,
                                                         (short)0, acc1, false, false);
        }
#pragma unroll
        for (int r = 0; r < 8; ++r) {
            const int row = r + half*8;                  // C/D: lanes>=16 hold M=r+8
            sS[w][row][k]      = acc0[r];                // logits cols 0..15
            sS[w][row][16 + k] = acc1[r];                // logits cols 16..31
        }
    }
#else
    {   // scalar fallback: lane owns one output column
        const int c = lane;
#pragma unroll
        for (int r = 0; r < KKN; ++r) {
            float v = 0.f;
#pragma unroll
            for (int u = 0; u < DOUTC; ++u) v = fmaf(sX[w][r][u], sWs[c*DOUTC+u], v);
            sS[w][r][c] = v;
        }
    }
#endif

    // ---- step 3: softmax over K + weighted sum, lane owns channel c ----
    {
        const int c = lane;
        float m = -3.4e38f;
#pragma unroll
        for (int r = 0; r < KKN; ++r) m = fmaxf(m, sS[w][r][c]);
        float Z = 0.f, f = 0.f;
#pragma unroll
        for (int r = 0; r < KKN; ++r) {
            const float e = expf(sS[w][r][c] - m);
            Z += e;
            f = fmaf(e, sX[w][r][c], f);
        }
        sF[w][c] = f / Z;
    }

    // ---- step 4: fpre = f @ Wp^T + bp (pre-BN, stats done later) ----
    if (lane < Cp) {
        float v = sPar[4*DH + lane];
#pragma unroll
        for (int u = 0; u < DOUTC; ++u) v = fmaf(sF[w][u], sWp[lane*DOUTC+u], v);
        fpre[(size_t)pt*Cp + lane] = v;
    }
}

// ======================================================================
// Kernel 6: mlp2 + BN(shortcut) + leaky 0.01 + transpose to (B,64,N,1)
// ======================================================================
__global__ __launch_bounds__(256) void final_kernel(const float* __restrict__ x3,
        const float* __restrict__ W2,    const float* __restrict__ b2,
        const float* __restrict__ scpre, const float* __restrict__ mvsc,
        const float* __restrict__ gsc,   const float* __restrict__ bscb,
        float* __restrict__ out) {
    const int p = blockIdx.x * 256 + threadIdx.x;
    const int b = p / NNP, n = p % NNP;
    float x[DOUTC];
#pragma unroll
    for (int u = 0; u < DOUTC; ++u) x[u] = x3[(size_t)p*DOUTC + u];
    for (int c = 0; c < 2*DOUTC; ++c) {
        float v = b2[c];
#pragma unroll
        for (int u = 0; u < DOUTC; ++u) v = fmaf(x[u], W2[c*DOUTC+u], v);
        float s = scpre[(size_t)p*(2*DOUTC) + c];
        s = (s - mvsc[c]) * mvsc[2*DOUTC + c] * gsc[c] + bscb[c];
        v += s;
        out[((size_t)b*(2*DOUTC) + c)*NNP + n] = v > 0.f ? v : 0.01f * v;
    }
}

// ======================================================================
extern "C" void kernel_launch(void* const* d_in, const int* in_sizes, int n_in,
                              void* d_out, int out_size, void* d_ws, size_t ws_size,
                              hipStream_t stream) {
    (void)in_sizes; (void)n_in; (void)out_size; (void)ws_size;
    const float* coords = (const float*)d_in[0];
    const float* feats  = (const float*)d_in[1];
    const float* W1  = (const float*)d_in[2];  const float* b1  = (const float*)d_in[3];
    const float* Wl1 = (const float*)d_in[4];  const float* bl1 = (const float*)d_in[5];
    const float* gl1 = (const float*)d_in[6];  const float* bl1b= (const float*)d_in[7];
    const float* Ws1 = (const float*)d_in[8];
    const float* Wp1 = (const float*)d_in[9];  const float* bp1 = (const float*)d_in[10];
    const float* gp1 = (const float*)d_in[11]; const float* bp1b= (const float*)d_in[12];
    const float* Wl2 = (const float*)d_in[13]; const float* bl2 = (const float*)d_in[14];
    const float* gl2 = (const float*)d_in[15]; const float* bl2b= (const float*)d_in[16];
    const float* Ws2 = (const float*)d_in[17];
    const float* Wp2 = (const float*)d_in[18]; const float* bp2 = (const float*)d_in[19];
    const float* gp2 = (const float*)d_in[20]; const float* bp2b= (const float*)d_in[21];
    const float* W2  = (const float*)d_in[22]; const float* b2  = (const float*)d_in[23];
    const float* Wsc = (const float*)d_in[24]; const float* bsc = (const float*)d_in[25];
    const float* gsc = (const float*)d_in[26]; const float* bscb= (const float*)d_in[27];
    float* out = (float*)d_out;

    // ---- workspace layout (4-byte units) ----
    float* ws = (float*)d_ws;
    size_t o = 0;
    int*   idx    = (int*)(ws + o); o += (size_t)NPTS * KKN;      // 1 M
    float* x1     = ws + o;         o += (size_t)NPTS * DH;       // 1 M
    float* x2     = ws + o;         o += (size_t)NPTS * DH;       // 1 M
    float* x3     = ws + o;         o += (size_t)NPTS * DOUTC;    // 2 M
    float* scpre  = ws + o;         o += (size_t)NPTS * 2*DOUTC;  // 4 M
    float* f1pre  = ws + o;         o += (size_t)NPTS * DH;       // 1 M
    float* f2pre  = ws + o;         o += (size_t)NPTS * DOUTC;    // 2 M
    float* part   = ws + o;         o += (size_t)LSG * 2 * 64;    // 128 K
    float* mv_l1  = ws + o;         o += 32;
    float* mv_p1  = ws + o;         o += 32;
    float* mv_l2  = ws + o;         o += 32;
    float* mv_p2  = ws + o;         o += 64;
    float* mv_sc  = ws + o;         o += 128;

    const int SG = 256;   // stats grid for bn_stats_mat

    knn_kernel<<<NPTS/256, 256, 0, stream>>>(coords, idx);
    mlp1_sc_kernel<<<NPTS/256, 256, 0, stream>>>(feats, W1, b1, Wsc, bsc, x1, scpre);

    // ---- stage 1: LSE1 BN stats -> fused lse+attpool -> BN(fpre)+relu ----
    lse_stats_kernel<<<LSG, 256, 0, stream>>>(coords, idx, Wl1, bl1, part);
    bn_finalize<<<1, 64, 0, stream>>>(part, LSG, (float)(NPTS*KKN), DH, mv_l1);
    attpool_kernel<<<NPTS/8, 256, 0, stream>>>(coords, idx, x1, Wl1, bl1, mv_l1,
                                               gl1, bl1b, Ws1, Wp1, bp1, DH, f1pre);
    bn_stats_mat<<<SG, 256, 0, stream>>>(f1pre, NPTS, DH, part);
    bn_finalize<<<1, 64, 0, stream>>>(part, SG, (float)NPTS, DH, mv_p1);
    bn_relu_kernel<<<(NPTS*DH)/256, 256, 0, stream>>>(f1pre, mv_p1, gp1, bp1b,
                                                      NPTS, DH, x2);

    // ---- stage 2 ----
    lse_stats_kernel<<<LSG, 256, 0, stream>>>(coords, idx, Wl2, bl2, part);
    bn_finalize<<<1, 64, 0, stream>>>(part, LSG, (float)(NPTS*KKN), DH, mv_l2);
    attpool_kernel<<<NPTS/8, 256, 0, stream>>>(coords, idx, x2, Wl2, bl2, mv_l2,
                                               gl2, bl2b, Ws2, Wp2, bp2, DOUTC, f2pre);
    bn_stats_mat<<<SG, 256, 0, stream>>>(f2pre, NPTS, DOUTC, part);
    bn_finalize<<<1, 64, 0, stream>>>(part, SG, (float)NPTS, DOUTC, mv_p2);
    bn_relu_kernel<<<(NPTS*DOUTC)/256, 256, 0, stream>>>(f2pre, mv_p2, gp2, bp2b,
                                                         NPTS, DOUTC, x3);

    // ---- shortcut BN stats + final fuse ----
    bn_stats_mat<<<SG, 256, 0, stream>>>(scpre, NPTS, 2*DOUTC, part);
    bn_finalize<<<1, 64, 0, stream>>>(part, SG, (float)NPTS, 2*DOUTC, mv_sc);
    final_kernel<<<NPTS/256, 256, 0, stream>>>(x3, W2, b2, scpre, mv_sc, gsc, bscb, out);
}